// LinearDML_35476429866020
// MI455X (gfx1250) — compile-verified
//
#include <hip/hip_runtime.h>

#define EPS 1e-6f
#define N_ROWS 200
#define D2 18
#define WAVES 8
#define STEPS_PER_WAVE 25

typedef float v2f __attribute__((ext_vector_type(2)));
typedef float v8f __attribute__((ext_vector_type(8)));

// Uniform-lane broadcast: v_readlane_b32 (SGPR result) instead of ds_bpermute.
__device__ __forceinline__ float lane_bcast(float v, int srcLane) {
  return __uint_as_float(__builtin_amdgcn_readlane(__float_as_uint(v), srcLane));
}

// One block per batch. 8 waves/block, each wave owns 25 consecutive prefix steps.
// Base Gram prefix built with V_WMMA_F32_16X16X4_F32; per-step rank-1 scan +
// wave-parallel 16x16 Gaussian solve (rows in lanes) for the two RHS.
__global__ __launch_bounds__(256) void LinearDML_kernel(const float* __restrict__ xtys,
                                                        float* __restrict__ out) {
  __shared__ float sx[N_ROWS * D2];       // 200x18 tile of this batch (14.4 KB)
  __shared__ float gstage[WAVES * 256];   // per-wave 16x16 WMMA->rows staging

  const int b   = blockIdx.x;
  const int tid = threadIdx.x;
  const float* src = xtys + b * (N_ROWS * D2);

  for (int idx = tid; idx < N_ROWS * D2; idx += 256)
    sx[idx] = src[idx];
  __syncthreads();

  const int lane = tid & 31;
  const int half = lane >> 4;
  const int r    = lane & 15;              // row/column index owned by this lane
  // wave-uniform scalars pinned to SGPRs -> scalar loop control, EXEC stays all-1
  const int wv    = __builtin_amdgcn_readfirstlane(tid >> 5);
  const int i0    = STEPS_PER_WAVE * wv;   // first prefix step for this wave
  const int nfull = i0 & ~3;               // multiple-of-4 rows covered by WMMA

  // ---------------- base Gram via f32 WMMA, 4 rows (K=4) per issue ----------
  // A (16x4, col k = x_{j+k}) and B (4x16, row k = x_{j+k}) share identical
  // register contents: reg v at lane-half h holds x_{j+2h+v}[lane%16].
  v8f acc = {0.f, 0.f, 0.f, 0.f, 0.f, 0.f, 0.f, 0.f};
  for (int j = 0; j < nfull; j += 4) {
    v2f a;
    a.x = sx[(j + 2 * half    ) * D2 + r];
    a.y = sx[(j + 2 * half + 1) * D2 + r];
    acc = __builtin_amdgcn_wmma_f32_16x16x4_f32(false, a, false, a,
                                                (short)0, acc, false, false);
  }

  // WMMA C-layout (vgpr v: M = v + 8*half, N = lane%16) -> row-per-lane
  float* gst = gstage + (wv << 8);
#pragma unroll
  for (int v = 0; v < 8; ++v)
    gst[(v + 8 * half) * 16 + r] = acc[v];
  __threadfence_block();                   // same-wave LDS is in-order; fence compiler

  float G[16];
#pragma unroll
  for (int c = 0; c < 16; ++c)
    G[c] = gst[r * 16 + c];

  // remainder rows nfull..i0-1: scalar rank-1 updates
  for (int j = nfull; j < i0; ++j) {
    float xr = sx[j * D2 + r];
#pragma unroll
    for (int c = 0; c < 16; ++c)
      G[c] = fmaf(xr, lane_bcast(xr, c), G[c]);
  }

  // base c_y, c_t, S_ty, S_tt over j < i0  (y = col 17, t = col 16)
  float cy = 0.f, ct = 0.f, sty = 0.f, stt = 0.f;
  for (int j = 0; j < i0; ++j) {
    float xr = sx[j * D2 + r];
    float tv = sx[j * D2 + 16];
    float yv = sx[j * D2 + 17];
    cy  = fmaf(xr, yv, cy);
    ct  = fmaf(xr, tv, ct);
    sty = fmaf(yv, tv, sty);
    stt = fmaf(tv, tv, stt);
  }

  // ---------------- 25 sequential prefix steps ------------------------------
#pragma unroll 1
  for (int s = 0; s < STEPS_PER_WAVE; ++s) {
    const int i = i0 + s;

    // augmented row [ G[r][:] + eps*e_r | cy[r] | ct[r] ]
    float ar[18];
#pragma unroll
    for (int c = 0; c < 16; ++c)
      ar[c] = G[c] + ((c == r) ? EPS : 0.f);
    ar[16] = cy;
    ar[17] = ct;

    // forward elimination (SPD + ridge, no pivoting); rows live in lanes
#pragma unroll
    for (int k = 0; k < 16; ++k) {
      float pivot = lane_bcast(ar[k], k);
      float fac   = ar[k] / pivot;
      bool below  = (r > k);
#pragma unroll
      for (int c = k + 1; c < 18; ++c) {
        float rv = lane_bcast(ar[c], k);
        float nv = fmaf(-fac, rv, ar[c]);
        ar[c] = below ? nv : ar[c];
      }
    }

    // back substitution for both RHS + on-the-fly dot products:
    //   dyct = w_y.c_t, dtct = w_t.c_t, dywt = w_y.w_t, dtwt = |w_t|^2
    float dyct = 0.f, dtct = 0.f, dywt = 0.f, dtwt = 0.f;
#pragma unroll
    for (int k = 15; k >= 0; --k) {
      float inv = 1.0f / lane_bcast(ar[k], k);
      float wyk = lane_bcast(ar[16], k) * inv;
      float wtk = lane_bcast(ar[17], k) * inv;
      bool above = (r < k);
      float akr = ar[k];
      float ny = fmaf(-akr, wyk, ar[16]);
      float nt = fmaf(-akr, wtk, ar[17]);
      ar[16] = above ? ny : ar[16];
      ar[17] = above ? nt : ar[17];
      float ctk = lane_bcast(ct, k);
      dyct = fmaf(wyk, ctk, dyct);
      dtct = fmaf(wtk, ctk, dtct);
      dywt = fmaf(wyk, wtk, dywt);
      dtwt = fmaf(wtk, wtk, dtwt);
    }

    // pred_i = (S_ty - w_y.c_t - eps*(w_y.w_t)) / (S_tt - w_t.c_t - eps*|w_t|^2 + eps)
    float num = sty - dyct - EPS * dywt;
    float den = stt - dtct - EPS * dtwt + EPS;
    if (lane == 0)
      out[b * N_ROWS + i] = num / den;     // preds.T -> out[b][i]

    // advance prefix with row i (rank-1 Gram update + vector/scalar sums)
    float xr = sx[i * D2 + r];
    float tv = sx[i * D2 + 16];
    float yv = sx[i * D2 + 17];
#pragma unroll
    for (int c = 0; c < 16; ++c)
      G[c] = fmaf(xr, lane_bcast(xr, c), G[c]);
    cy  = fmaf(xr, yv, cy);
    ct  = fmaf(xr, tv, ct);
    sty = fmaf(yv, tv, sty);
    stt = fmaf(tv, tv, stt);
  }
}

extern "C" void kernel_launch(void* const* d_in, const int* in_sizes, int n_in,
                              void* d_out, int out_size, void* d_ws, size_t ws_size,
                              hipStream_t stream) {
  (void)n_in; (void)out_size; (void)d_ws; (void)ws_size;
  const float* xtys = (const float*)d_in[0];
  float* out = (float*)d_out;
  const int nb = in_sizes[0] / (N_ROWS * D2);   // 64 batches
  LinearDML_kernel<<<nb, 256, 0, stream>>>(xtys, out);
}